// SAGE_73237782332047
// MI455X (gfx1250) — compile-verified
//
#include <hip/hip_runtime.h>

typedef __attribute__((ext_vector_type(2))) float v2f;
typedef __attribute__((ext_vector_type(4))) float v4f;
typedef __attribute__((ext_vector_type(8))) float v8f;

#define DIM 128
#define LAYERS 4

// ---------------- degree kernels ----------------
__global__ __launch_bounds__(256) void k_deg(const int* __restrict__ dst,
                                             float* __restrict__ deg, int E) {
    int e = blockIdx.x * 256 + threadIdx.x;
    if (e < E) unsafeAtomicAdd(&deg[dst[e]], 1.0f);
}

__global__ __launch_bounds__(256) void k_invdeg(float* __restrict__ deg, int n) {
    int i = blockIdx.x * 256 + threadIdx.x;
    if (i < n) deg[i] = 1.0f / fmaxf(deg[i], 1.0f);
}

// ---------------- scatter: agg[dst,:] += h[src,:], one wave per edge ----------------
__global__ __launch_bounds__(256) void k_scatter(const float* __restrict__ h,
                                                 const int* __restrict__ src,
                                                 const int* __restrict__ dstv,
                                                 float* __restrict__ agg, int E) {
    int wave = (blockIdx.x * 256 + threadIdx.x) >> 5;
    int lane = threadIdx.x & 31;
    if (wave >= E) return;
    int s = src[wave];
    int d = dstv[wave];
    // 128 floats per row = 32 lanes x float4
    v4f v = *(reinterpret_cast<const v4f*>(h + (size_t)s * DIM) + lane);
    float* ap = agg + (size_t)d * DIM + lane * 4;
    unsafeAtomicAdd(ap + 0, v[0]);
    unsafeAtomicAdd(ap + 1, v[1]);
    unsafeAtomicAdd(ap + 2, v[2]);
    unsafeAtomicAdd(ap + 3, v[3]);
}

// ---------------- fused SAGE GEMM: hout = relu(agg*invdeg @ Wl + hin @ Wr + b) ----
// Block: 256 threads = 8 waves; block covers 128 rows x 128 cols.
// Wave w: rows [blockRow + 16w, +16), all 128 cols as 8 column tiles of 16.
// f32 WMMA 16x16x4; two clean K=128 loops (no per-iteration operand select).
__global__ __launch_bounds__(256) void k_gemm(const float* __restrict__ agg,
                                              const float* __restrict__ hin,
                                              const float* __restrict__ Wl,
                                              const float* __restrict__ Wr,
                                              const float* __restrict__ bias,
                                              const float* __restrict__ invdeg,
                                              float* __restrict__ hout, int n) {
    const int wave  = threadIdx.x >> 5;
    const int lane  = threadIdx.x & 31;
    const int n15   = lane & 15;
    const int khalf = lane >> 4;             // 0: K+0/K+1, 1: K+2/K+3  (ISA 16x4 f32 A layout)
    const int rowBase = blockIdx.x * 128 + wave * 16;

    int rowA = rowBase + n15;                // A-fragment row for this lane
    if (rowA >= n) rowA = n - 1;             // clamp: keeps loads in-bounds, EXEC stays full
    const float invd = invdeg[rowA];

    v8f acc[8] = {};                         // 8 column tiles, f32 C/D = 8 VGPRs each

    // ---- half 1: acc += (agg * inv_deg) @ Wl ----
    {
        const float* aRow = agg + (size_t)rowA * DIM + 2 * khalf;
        const float* bCol = Wl + (size_t)(2 * khalf) * DIM + n15;
        #pragma unroll 4
        for (int k0 = 0; k0 < DIM; k0 += 4) {
            v2f a;
            a.x = aRow[k0 + 0] * invd;
            a.y = aRow[k0 + 1] * invd;
            const float* bp = bCol + (size_t)k0 * DIM;
            #pragma unroll
            for (int t = 0; t < 8; ++t) {
                v2f bf;
                bf.x = bp[t * 16];
                bf.y = bp[t * 16 + DIM];
                acc[t] = __builtin_amdgcn_wmma_f32_16x16x4_f32(
                    false, a, false, bf, (short)0, acc[t], false, false);
            }
        }
    }

    // ---- half 2: acc += hin @ Wr ----
    {
        const float* aRow = hin + (size_t)rowA * DIM + 2 * khalf;
        const float* bCol = Wr + (size_t)(2 * khalf) * DIM + n15;
        #pragma unroll 4
        for (int k0 = 0; k0 < DIM; k0 += 4) {
            v2f a;
            a.x = aRow[k0 + 0];
            a.y = aRow[k0 + 1];
            const float* bp = bCol + (size_t)k0 * DIM;
            #pragma unroll
            for (int t = 0; t < 8; ++t) {
                v2f bf;
                bf.x = bp[t * 16];
                bf.y = bp[t * 16 + DIM];
                acc[t] = __builtin_amdgcn_wmma_f32_16x16x4_f32(
                    false, a, false, bf, (short)0, acc[t], false, false);
            }
        }
    }

    // Epilogue: bias + ReLU.  C/D layout: VGPR i -> row rowBase + i + 8*khalf, col = n15.
    #pragma unroll
    for (int t = 0; t < 8; ++t) {
        const int col = t * 16 + n15;
        const float bb = bias[col];
        #pragma unroll
        for (int i = 0; i < 8; ++i) {
            int r = rowBase + i + 8 * khalf;
            if (r < n) hout[(size_t)r * DIM + col] = fmaxf(acc[t][i] + bb, 0.0f);
        }
    }
}

extern "C" void kernel_launch(void* const* d_in, const int* in_sizes, int n_in,
                              void* d_out, int out_size, void* d_ws, size_t ws_size,
                              hipStream_t stream) {
    const float* x    = (const float*)d_in[0];
    const float* Wl   = (const float*)d_in[1];
    const float* Wr   = (const float*)d_in[2];
    const float* bias = (const float*)d_in[3];
    const int*   ei   = (const int*)d_in[4];    // [2, E]: row0 = src, row1 = dst

    const int n = in_sizes[0] / DIM;            // 50000
    const int E = in_sizes[4] / 2;              // 600000

    float* out = (float*)d_out;
    float* agg = (float*)d_ws;                  // [n, 128]
    float* hA  = agg + (size_t)n * DIM;         // [n, 128] ping buffer
    float* deg = hA + (size_t)n * DIM;          // [n] degree -> inv_deg (in place)

    // Degrees (computed once, reused by every layer)
    hipMemsetAsync(deg, 0, (size_t)n * sizeof(float), stream);
    k_deg<<<(E + 255) / 256, 256, 0, stream>>>(ei + E, deg, E);
    k_invdeg<<<(n + 255) / 256, 256, 0, stream>>>(deg, n);

    // Ping-pong: x -> hA -> out -> hA -> out  (last layer lands in d_out)
    const float* hin = x;
    for (int l = 0; l < LAYERS; ++l) {
        float* hout = ((LAYERS - 1 - l) & 1) ? hA : out;
        hipMemsetAsync(agg, 0, (size_t)n * DIM * sizeof(float), stream);
        k_scatter<<<(E + 7) / 8, 256, 0, stream>>>(hin, ei, ei + E, agg, E);
        k_gemm<<<(n + 127) / 128, 256, 0, stream>>>(
            agg, hin,
            Wl + (size_t)l * DIM * DIM,
            Wr + (size_t)l * DIM * DIM,
            bias + (size_t)l * DIM,
            deg, hout, n);
        hin = hout;
    }
}